// WeightedLabelSmoothCE_18047452578216
// MI455X (gfx1250) — compile-verified
//
#include <hip/hip_runtime.h>
#include <cstdint>

#define NCLASS     64
#define TILE_ROWS  64
#define TILE_ELEMS (TILE_ROWS * NCLASS)   // 4096 f32 = 16 KB per tile
#define BLOCK      64
#define GRID       4096

typedef __attribute__((ext_vector_type(4))) unsigned int u32x4;
typedef __attribute__((ext_vector_type(8))) int          i32x8;
typedef __attribute__((ext_vector_type(4))) int          i32x4;

// Tensor Data Mover: DMA one contiguous 1-D tile (TILE_ELEMS f32) from global
// memory into LDS at byte offset lds_off. Descriptor layout per CDNA5 ISA
// ch. 7/8 (group0: count/lds_addr/global_addr/type, group1: data_size + dims).
// This toolchain exposes the 6-arg builtin (g0, g1, g2, g3, extra, cpol).
__device__ __forceinline__ void tdm_load_tile(const float* gsrc, unsigned lds_off) {
  unsigned long long ga = (unsigned long long)(uintptr_t)gsrc;
  u32x4 g0;
  g0[0] = 1u;                                                  // count = 1 (valid D#)
  g0[1] = lds_off;                                             // lds_addr (bytes)
  g0[2] = (unsigned)(ga & 0xffffffffu);                        // global_addr[31:0]
  g0[3] = (unsigned)((ga >> 32) & 0x01ffffffu) | (2u << 30);   // addr[56:32] | type=2
  i32x8 g1;
  g1[0] = (int)(2u << 16);                       // data_size = 4B; no multicast/flags
  g1[1] = (int)((TILE_ELEMS & 0xffff) << 16);    // tensor_dim0[15:0]
  g1[2] = (int)(((TILE_ELEMS >> 16) & 0xffff) | (1u << 16)); // dim0[31:16]; tensor_dim1=1
  g1[3] = (int)((TILE_ELEMS & 0xffff) << 16);    // tensor_dim1[31:16]=0; tile_dim0
  g1[4] = 0;                                     // tile_dim1 = 0 (unused); tile_dim2 = 0
  g1[5] = (int)TILE_ELEMS;                       // tensor_dim0_stride[31:0]
  g1[6] = (int)((TILE_ELEMS & 0xffff) << 16);    // stride0[47:32]=0; stride1[15:0]
  g1[7] = 0;                                     // stride1[47:16] = 0
  i32x4 gz4 = {0, 0, 0, 0};                      // groups 2/3 unused (<=2D tensor)
  i32x8 gz8 = {0, 0, 0, 0, 0, 0, 0, 0};
  __builtin_amdgcn_tensor_load_to_lds(g0, g1, gz4, gz4, gz8, 0);
}

__global__ void __launch_bounds__(BLOCK)
ce_main(const float* __restrict__ x, const int* __restrict__ tgt,
        const float* __restrict__ cw, float* __restrict__ partial, int rows) {
  __shared__ float tile[2][TILE_ELEMS];
  __shared__ float lds_w[NCLASS];
  __shared__ float red[BLOCK];

  constexpr float S  = 0.05f;
  constexpr float Bc = S / (float)(NCLASS - 1);        // s/(C-1)
  constexpr float A  = (1.0f - S) - Bc;                // (1-s) - s/(C-1)

  const int tid = threadIdx.x;
  lds_w[tid] = cw[tid];                                // BLOCK == NCLASS == 64

  // Uniform class weights -> scalar loads (SGPRs); also total weight W.
  float4 wv[NCLASS / 4];
  float  W = 0.0f;
#pragma unroll
  for (int k = 0; k < NCLASS / 4; ++k) {
    wv[k] = ((const float4*)cw)[k];
    W += wv[k].x + wv[k].y + wv[k].z + wv[k].w;
  }

  const int ntiles = rows / TILE_ROWS;
  const unsigned lds_off0 = (unsigned)(uintptr_t)&tile[0][0];
  const unsigned lds_off1 = (unsigned)(uintptr_t)&tile[1][0];

  // Wave 0 drives the TDM pipeline: prefetch first tile into buffer 0.
  if (tid < 32 && (int)blockIdx.x < ntiles) {
    tdm_load_tile(x + (size_t)blockIdx.x * TILE_ELEMS, lds_off0);
  }

  float acc = 0.0f;
  int it = 0;
  for (int tIdx = blockIdx.x; tIdx < ntiles; tIdx += GRID, ++it) {
    const int buf = it & 1;
    if (tid < 32) {
      const int nextT = tIdx + GRID;
      if (nextT < ntiles) {   // prefetch next tile into the other buffer
        tdm_load_tile(x + (size_t)nextT * TILE_ELEMS, buf ? lds_off0 : lds_off1);
        __builtin_amdgcn_s_wait_tensorcnt(1);   // oldest (current tile) done
      } else {
        __builtin_amdgcn_s_wait_tensorcnt(0);
      }
    }
    __syncthreads();          // tile[buf] visible to all waves (+ lds_w on iter 0)

    const float* __restrict__ row = &tile[buf][tid * NCLASS];
    float4 v[NCLASS / 4];
#pragma unroll
    for (int k = 0; k < NCLASS / 4; ++k) v[k] = ((const float4*)row)[k];

    float m = v[0].x;
#pragma unroll
    for (int k = 0; k < NCLASS / 4; ++k) {
      m = fmaxf(m, v[k].x); m = fmaxf(m, v[k].y);
      m = fmaxf(m, v[k].z); m = fmaxf(m, v[k].w);
    }
    float se = 0.0f, wx = 0.0f;
#pragma unroll
    for (int k = 0; k < NCLASS / 4; ++k) {
      se += __expf(v[k].x - m) + __expf(v[k].y - m) +
            __expf(v[k].z - m) + __expf(v[k].w - m);
      wx = fmaf(v[k].x, wv[k].x, wx); wx = fmaf(v[k].y, wv[k].y, wx);
      wx = fmaf(v[k].z, wv[k].z, wx); wx = fmaf(v[k].w, wv[k].w, wx);
    }
    const float lse    = m + __logf(se);
    const int   rowIdx = tIdx * TILE_ROWS + tid;
    const int   t      = tgt[rowIdx];
    const float xt     = row[t];
    const float wt     = lds_w[t];
    acc += A * wt * (xt - lse) + Bc * (wx - W * lse);

    __syncthreads();          // everyone done with tile[buf] before it is reused
  }

  // Deterministic block tree reduction.
  red[tid] = acc;
  __syncthreads();
#pragma unroll
  for (int st = BLOCK / 2; st > 0; st >>= 1) {
    if (tid < st) red[tid] += red[tid + st];
    __syncthreads();
  }
  if (tid == 0) partial[blockIdx.x] = red[0];
}

__global__ void ce_reduce(const float* __restrict__ partial, float* __restrict__ out,
                          int n, float inv_rows) {
  __shared__ float red[256];
  float s = 0.0f;
  for (int i = threadIdx.x; i < n; i += 256) s += partial[i];
  red[threadIdx.x] = s;
  __syncthreads();
  for (int st = 128; st > 0; st >>= 1) {
    if ((int)threadIdx.x < st) red[threadIdx.x] += red[threadIdx.x + st];
    __syncthreads();
  }
  if (threadIdx.x == 0) out[0] = -red[0] * inv_rows;
}

extern "C" void kernel_launch(void* const* d_in, const int* in_sizes, int n_in,
                              void* d_out, int out_size, void* d_ws, size_t ws_size,
                              hipStream_t stream) {
  const float* x   = (const float*)d_in[0];   // [B,4,64] f32 logits
  const int*   tg  = (const int*)d_in[1];     // [B,4] int targets
  const float* cw  = (const float*)d_in[2];   // [64] f32 class weights
  float*       out = (float*)d_out;           // scalar loss
  float*       ws  = (float*)d_ws;            // GRID partial sums

  const int rows = in_sizes[0] / NCLASS;      // B*4 = 1,048,576

  ce_main<<<GRID, BLOCK, 0, stream>>>(x, tg, cw, ws, rows);
  ce_reduce<<<1, 256, 0, stream>>>(ws, out, GRID, 1.0f / (float)rows);
}